// dPLPM3_54228257079820
// MI455X (gfx1250) — compile-verified
//
#include <hip/hip_runtime.h>
#include <hip/hip_bf16.h>
#include <math.h>

typedef __attribute__((ext_vector_type(2))) float v2f;
typedef __attribute__((ext_vector_type(8))) float v8f;

#define B_     4
#define NB_    8
#define FB_    32
#define T_     60000
#define TILE_T 256
#define ROWLEN (TILE_T + 1)   // 257 -> row stride odd, no LDS bank clash across rows

// ---------------------------------------------------------------------------
// K1: x[b, band*32+f, t] -> xp[b,band,t] = sum_f relu(log1p(g*x[t+1]) - log1p(g*x[t]) + diff_b)
// Heavy streaming kernel. Frequency pooling done on the matrix unit:
// V_WMMA_F32_16X16X4_F32 with B = all-ones (D[m,n] = rowsum(A), layout-proof).
// ---------------------------------------------------------------------------
__global__ __launch_bounds__(256) void k1_pool(
    const float* __restrict__ x, const float* __restrict__ log_gamma,
    const float* __restrict__ diff_b, float* __restrict__ xp)
{
  __shared__ float sl[FB_ * ROWLEN];

  const int tile = blockIdx.x;
  const int bc   = blockIdx.y;            // b*NB + band
  const int band = bc & (NB_ - 1);
  const int t0   = tile * TILE_T;

  const float gamma = expf(log_gamma[band]);
  const float db    = diff_b[band];
  // (b*256 + band*32 + f)*T = (bc*32 + f)*T
  const float* xb = x + (size_t)bc * FB_ * T_;

  // Stage tile [32 f][257 t] as log1p(gamma*x); zero-pad (log1p(0)==0 matches conv pad).
  for (int idx = threadIdx.x; idx < FB_ * ROWLEN; idx += 256) {
    const int f  = idx / ROWLEN;
    const int tt = idx - f * ROWLEN;
    const int t  = t0 + tt;
    float v = 0.0f;
    if (t < T_) v = __builtin_nontemporal_load(xb + (size_t)f * T_ + t);
    sl[f * ROWLEN + tt] = log1pf(gamma * v);
  }
  __syncthreads();

  const int lane = threadIdx.x & 31;
  const int wave = threadIdx.x >> 5;      // 8 waves
  const int m    = lane & 15;             // t within 16-group (A: M = lane&15)
  const int kh   = lane >> 4;             // A: lanes 16-31 carry K=2,3
  float* xp_out = xp + (size_t)bc * T_;

  for (int g = 0; g < 2; ++g) {           // each wave: two 16-t groups -> 8*2*16 = 256 t
    const int tb = (wave * 2 + g) * 16;
    v8f c = {0.f,0.f,0.f,0.f,0.f,0.f,0.f,0.f};
    const v2f ones = {1.f, 1.f};
    #pragma unroll
    for (int fc4 = 0; fc4 < 8; ++fc4) {   // K-chunks of 4 over f = 0..31
      const int f0 = fc4 * 4 + kh * 2;
      const float* r0 = &sl[(f0    ) * ROWLEN + tb + m];
      const float* r1 = &sl[(f0 + 1) * ROWLEN + tb + m];
      v2f a;
      a.x = fmaxf(r0[1] - r0[0] + db, 0.0f);   // K = f0
      a.y = fmaxf(r1[1] - r1[0] + db, 0.0f);   // K = f0+1
      c = __builtin_amdgcn_wmma_f32_16x16x4_f32(false, a, false, ones,
                                                (short)0, c, false, false);
    }
    // D layout: lane 0 holds D[M=0..7,N=0] = xp[tb..tb+7]; lane 16 holds M=8..15.
    if ((lane & 15) == 0) {
      const int ts = t0 + tb + kh * 8;    // T % 16 == 0 -> whole-group validity
      if (ts < T_) {
        *reinterpret_cast<float4*>(xp_out + ts)     = make_float4(c[0], c[1], c[2], c[3]);
        *reinterpret_cast<float4*>(xp_out + ts + 4) = make_float4(c[4], c[5], c[6], c[7]);
      }
    }
  }
}

// ---------------------------------------------------------------------------
// K2: sm[b,t] = sum_c mix_w[c] * xp[b,c,t]
// ---------------------------------------------------------------------------
__global__ __launch_bounds__(256) void k2_mix(
    const float* __restrict__ xp, const float* __restrict__ mix_w,
    float* __restrict__ sm)
{
  const int t = blockIdx.x * 256 + threadIdx.x;
  const int b = blockIdx.y;
  if (t >= T_) return;
  float acc = 0.f;
  #pragma unroll
  for (int c = 0; c < NB_; ++c)
    acc += mix_w[c] * xp[((size_t)b * NB_ + c) * T_ + t];
  sm[(size_t)b * T_ + t] = acc;
}

// ---------------------------------------------------------------------------
// K3: xm[t] = sm[t] - (la_w0 * sum_{j=-5..5,in-range} sm[t+j] + sum_c mix_w[c]*la_b[c])
// ---------------------------------------------------------------------------
__global__ __launch_bounds__(256) void k3_localavg(
    const float* __restrict__ sm, const float* __restrict__ la_w,
    const float* __restrict__ la_b, const float* __restrict__ mix_w,
    float* __restrict__ xm)
{
  const int t = blockIdx.x * 256 + threadIdx.x;
  const int b = blockIdx.y;
  if (t >= T_) return;
  const float lw = la_w[0];
  float lab = 0.f;
  #pragma unroll
  for (int c = 0; c < NB_; ++c) lab += mix_w[c] * la_b[c];
  const float* smb = sm + (size_t)b * T_;
  float acc = 0.f;
  #pragma unroll
  for (int j = -5; j <= 5; ++j) {
    const int tt = t + j;
    if (tt >= 0 && tt < T_) acc += smb[tt];
  }
  xm[(size_t)b * T_ + t] = smb[t] - (lw * acc + lab);
}

// ---------------------------------------------------------------------------
// K4: xg[t] = relu(sum_k g_w[k]*xm[t+k-7] + g_b); per-batch max via int-bits atomicMax
// ---------------------------------------------------------------------------
__global__ __launch_bounds__(256) void k4_gauss_max(
    const float* __restrict__ xm, const float* __restrict__ g_w,
    const float* __restrict__ g_b, float* __restrict__ xg, int* __restrict__ mg)
{
  __shared__ float red[256];
  const int t = blockIdx.x * 256 + threadIdx.x;
  const int b = blockIdx.y;
  float v = 0.f;
  if (t < T_) {
    float acc = g_b[0];
    const float* xmb = xm + (size_t)b * T_;
    #pragma unroll
    for (int k = 0; k < 15; ++k) {
      const int tt = t + k - 7;
      if (tt >= 0 && tt < T_) acc += g_w[k] * xmb[tt];
    }
    v = fmaxf(acc, 0.f);
    xg[(size_t)b * T_ + t] = v;
  }
  red[threadIdx.x] = v;
  __syncthreads();
  for (int s = 128; s > 0; s >>= 1) {
    if (threadIdx.x < s) red[threadIdx.x] = fmaxf(red[threadIdx.x], red[threadIdx.x + s]);
    __syncthreads();
  }
  if (threadIdx.x == 0) atomicMax(mg + b, __float_as_int(red[0])); // v >= 0
}

// ---------------------------------------------------------------------------
// K5: spectral_flux = xg / (max + EPS)  -> d_out[0 : B*T]
// ---------------------------------------------------------------------------
__global__ __launch_bounds__(256) void k5_norm(
    const float* __restrict__ xg, const int* __restrict__ mg, float* __restrict__ out0)
{
  const int t = blockIdx.x * 256 + threadIdx.x;
  const int b = blockIdx.y;
  if (t >= T_) return;
  const float mx = __int_as_float(mg[b]);
  out0[(size_t)b * T_ + t] = xg[(size_t)b * T_ + t] / (mx + 1e-8f);
}

// ---------------------------------------------------------------------------
// K6: fa[t] = sigmoid(sum_k fg_w[k]*((sum fc_w)*sf[t+k-7] + fc_b) + fg_b); max -> mfa
// ---------------------------------------------------------------------------
__global__ __launch_bounds__(256) void k6_fuse_max(
    const float* __restrict__ sf, const float* __restrict__ fc_w,
    const float* __restrict__ fc_b, const float* __restrict__ fg_w,
    const float* __restrict__ fg_b, float* __restrict__ fa, int* __restrict__ mfa)
{
  __shared__ float red[256];
  const int t = blockIdx.x * 256 + threadIdx.x;
  const int b = blockIdx.y;
  float v = 0.f;
  if (t < T_) {
    const float s4  = fc_w[0] + fc_w[1] + fc_w[2] + fc_w[3];
    const float fcb = fc_b[0];
    float acc = fg_b[0];
    const float* sfb = sf + (size_t)b * T_;
    #pragma unroll
    for (int k = 0; k < 15; ++k) {
      const int tt = t + k - 7;
      if (tt >= 0 && tt < T_) acc += fg_w[k] * (s4 * sfb[tt] + fcb);
    }
    v = 1.0f / (1.0f + expf(-acc));
    fa[(size_t)b * T_ + t] = v;
  }
  red[threadIdx.x] = v;
  __syncthreads();
  for (int s = 128; s > 0; s >>= 1) {
    if (threadIdx.x < s) red[threadIdx.x] = fmaxf(red[threadIdx.x], red[threadIdx.x + s]);
    __syncthreads();
  }
  if (threadIdx.x == 0) atomicMax(mfa + b, __float_as_int(red[0])); // v in (0,1)
}

// ---------------------------------------------------------------------------
// K7: fused_nov = fa / (max + EPS) -> d_out[B*T : 2*B*T]
// ---------------------------------------------------------------------------
__global__ __launch_bounds__(256) void k7_norm2(
    const float* __restrict__ fa, const int* __restrict__ mfa, float* __restrict__ out1)
{
  const int t = blockIdx.x * 256 + threadIdx.x;
  const int b = blockIdx.y;
  if (t >= T_) return;
  const float mx = __int_as_float(mfa[b]);
  out1[(size_t)b * T_ + t] = fa[(size_t)b * T_ + t] / (mx + 1e-8f);
}

__global__ void k0_init(int* __restrict__ mg)
{
  if (threadIdx.x < 8) mg[threadIdx.x] = 0;  // mg[0..3], mfa[0..3]
}

extern "C" void kernel_launch(void* const* d_in, const int* in_sizes, int n_in,
                              void* d_out, int out_size, void* d_ws, size_t ws_size,
                              hipStream_t stream) {
  const float* x         = (const float*)d_in[0];
  const float* log_gamma = (const float*)d_in[1];
  /* d_in[2] diff_w: structural [-1,+1] taps, folded into k1 */
  const float* diff_b    = (const float*)d_in[3];
  const float* la_w      = (const float*)d_in[4];
  const float* la_b      = (const float*)d_in[5];
  const float* mix_w     = (const float*)d_in[6];
  const float* g_w       = (const float*)d_in[7];
  const float* g_b       = (const float*)d_in[8];
  const float* fc_w      = (const float*)d_in[9];
  const float* fc_b      = (const float*)d_in[10];
  const float* fg_w      = (const float*)d_in[11];
  const float* fg_b      = (const float*)d_in[12];
  float* out = (float*)d_out;

  // Workspace layout (floats): xp[1.92M] sm[240k] xm[240k] xg[240k] fa[240k] + 8 ints
  float* xp = (float*)d_ws;
  float* sm = xp + (size_t)B_ * NB_ * T_;
  float* xm = sm + (size_t)B_ * T_;
  float* xg = xm + (size_t)B_ * T_;
  float* fa = xg + (size_t)B_ * T_;
  int*   mg = (int*)(fa + (size_t)B_ * T_);   // mg[0..3], mfa = mg+4

  const dim3 g1((T_ + TILE_T - 1) / TILE_T, B_ * NB_);
  const dim3 g2((T_ + 255) / 256, B_);

  k1_pool    <<<g1, 256, 0, stream>>>(x, log_gamma, diff_b, xp);
  k2_mix     <<<g2, 256, 0, stream>>>(xp, mix_w, sm);
  k3_localavg<<<g2, 256, 0, stream>>>(sm, la_w, la_b, mix_w, xm);
  k0_init    <<<1,   8,  0, stream>>>(mg);
  k4_gauss_max<<<g2, 256, 0, stream>>>(xm, g_w, g_b, xg, mg);
  k5_norm    <<<g2, 256, 0, stream>>>(xg, mg, out);
  k6_fuse_max<<<g2, 256, 0, stream>>>(out, fc_w, fc_b, fg_w, fg_b, fa, mg + 4);
  k7_norm2   <<<g2, 256, 0, stream>>>(fa, mg + 4, out + (size_t)B_ * T_);
}